// CombinedICIRLoss_73057393705017
// MI455X (gfx1250) — compile-verified
//
#include <hip/hip_runtime.h>
#include <hip/hip_bf16.h>
#include <stdint.h>

// ---------------------------------------------------------------------------
// CombinedICIRLoss for MI455X (gfx1250, wave32).
//   B=32, N=1024.  O(B*N^2) pair work = f32 VALU + v_tanh_f32 (TRANS); not
//   expressible as matmul, so WMMA is used for wave32 cross-lane reductions.
//   Rows staged into LDS by the Tensor Data Mover; pair loop reads interleaved
//   float2 {p,t} from LDS (one ds_load_b64/iter) and is fully branchless.
// ---------------------------------------------------------------------------

#define N_ITEMS   1024
#define N_BATCH   32
#define N_SLICES  8          // j-slice blocks per batch -> 256 blocks total
#define NT_MAIN   64         // 2 waves per block
#define NWAVES    (NT_MAIN / 32)
#define M_ROWS    (N_ITEMS / N_SLICES)   // 128 rows per slice

typedef __attribute__((ext_vector_type(2))) float        v2f;
typedef __attribute__((ext_vector_type(8))) float        v8f;
typedef __attribute__((ext_vector_type(4))) unsigned int v4u;
typedef __attribute__((ext_vector_type(4))) int          v4i;
typedef __attribute__((ext_vector_type(8))) int          v8i;

// ----------------------------- fast tanh (TRANS unit) ----------------------
__device__ __forceinline__ float fast_tanh(float x) {
#if __has_builtin(__builtin_amdgcn_tanhf)
    return __builtin_amdgcn_tanhf(x);          // v_tanh_f32 on gfx1250
#else
    float e = __expf(x + x);                   // tanh(x) = 1 - 2/(e^{2x}+1)
    return 1.0f - 2.0f / (e + 1.0f);
#endif
}

// ----------------------------- wave32 reductions ---------------------------
__device__ __forceinline__ float wave_sum(float v) {
    for (int o = 16; o > 0; o >>= 1) v += __shfl_xor(v, o, 32);
    return v;
}
__device__ __forceinline__ float wave_max(float v) {
    for (int o = 16; o > 0; o >>= 1) v = fmaxf(v, __shfl_xor(v, o, 32));
    return v;
}

// Full-wave sum via V_WMMA_F32_16X16X4_F32 with a ones B-matrix.
// A layout (16x4 f32): lane L<16 -> row L (K0,K1); lane L>=16 -> row L-16 (K2,K3).
// With A = {x,0} per lane and ones B: D[m,n] = x_m + x_{m+16}.  Summing the 8
// D VGPRs yields rows 0..7 (lanes 0-15) / 8..15 (lanes 16-31); one xor-16
// shuffle completes the 32-lane sum in every lane.
__device__ __forceinline__ float wave_sum_wmma(float x) {
#if __has_builtin(__builtin_amdgcn_wmma_f32_16x16x4_f32)
    v2f a; a.x = x;    a.y = 0.0f;
    v2f b; b.x = 1.0f; b.y = 1.0f;
    v8f c = {};
    c = __builtin_amdgcn_wmma_f32_16x16x4_f32(false, a, false, b,
                                              (short)0, c, false, false);
    float s = ((c[0] + c[1]) + (c[2] + c[3])) + ((c[4] + c[5]) + (c[6] + c[7]));
    s += __shfl_xor(s, 16, 32);
    return s;
#else
    return wave_sum(x);
#endif
}

// ----------------------------- block reductions ----------------------------
__device__ __forceinline__ float block_sum(float v, float* sc, int tid) {
    float w = wave_sum(v);
    __syncthreads();
    if ((tid & 31) == 0) sc[tid >> 5] = w;
    __syncthreads();
    float r = 0.0f;
#pragma unroll
    for (int k = 0; k < NWAVES; ++k) r += sc[k];
    return r;
}
__device__ __forceinline__ float block_max(float v, float* sc, int tid) {
    float w = wave_max(v);
    __syncthreads();
    if ((tid & 31) == 0) sc[tid >> 5] = w;
    __syncthreads();
    float r = sc[0];
#pragma unroll
    for (int k = 1; k < NWAVES; ++k) r = fmaxf(r, sc[k]);
    return r;
}

// ----------------------------- TDM staging ---------------------------------
#if __has_builtin(__builtin_amdgcn_tensor_load_to_lds)
#define HAVE_TDM 1
// D# for a contiguous row of `nelem` f32: data_size=4B,
// tensor_dim0 = tile_dim0 = nelem, tensor_dim1 = tile_dim1 = 1, type=2.
__device__ __forceinline__ void tdm_load_row_f32(const float* gsrc,
                                                 unsigned lds_byte_off,
                                                 int nelem) {
    unsigned long long ga = (unsigned long long)(uintptr_t)gsrc;
    v4u g0 = { 1u,                                     // count=1 (valid), user desc
               lds_byte_off,                           // lds_addr [63:32]
               (unsigned)(ga & 0xFFFFFFFFull),         // global_addr lo
               (unsigned)((ga >> 32) & 0x01FFFFFFull)  // global_addr hi
                   | (2u << 30) };                     // type=2 ("image")
    v8i g1 = { (int)(2u << 16),                        // data_size=2 -> 4 bytes
               (int)((unsigned)nelem << 16),           // tensor_dim0[15:0] @ [63:48]
               (int)(1u << 16),                        // tensor_dim1=1   @ [111:80]
               (int)((unsigned)nelem << 16),           // tile_dim0       @ [127:112]
               1,                                      // tile_dim1=1     @ [143:128]
               nelem,                                  // tensor_dim0_stride lo
               0, 0 };
    v4i gz = { 0, 0, 0, 0 };
#if __clang_major__ >= 23
    v8i gz8 = { 0, 0, 0, 0, 0, 0, 0, 0 };
    __builtin_amdgcn_tensor_load_to_lds(g0, g1, gz, gz, gz8, 0);
#else
    __builtin_amdgcn_tensor_load_to_lds(g0, g1, gz, gz, 0);
#endif
}
#endif

// ----------------------------- main kernel ---------------------------------
struct SMem {
    float  p[N_ITEMS];        // TDM landing zone (pred row)
    float  t[N_ITEMS];        // TDM landing zone (target row)
    float2 pt[N_ITEMS];       // interleaved {p, t} for the pair loop
    float  sc0[NWAVES];
    float  sc1[NWAVES];
};

// grid = (N_BATCH, N_SLICES); block = NT_MAIN (2 waves).
// Per (batch b, slice s): rows i = s + 8*m, m in [0,128).
// Thread tid owns m = tid and m = 127 - tid -> constant per-thread work.
// Slice 0 additionally computes ListNet KL and n_valid.
// ws[(s*N_BATCH + b)*4 + {0,1,2,3}] = {kendall_num, pw_num, kl, n_valid}.
__global__ __launch_bounds__(NT_MAIN)
void combined_loss_main(const float* __restrict__ pred,
                        const float* __restrict__ tgt,
                        float* __restrict__ ws) {
    __shared__ SMem sm;
    const int b   = blockIdx.x;
    const int s   = blockIdx.y;
    const int tid = threadIdx.x;
    const float* gp = pred + b * N_ITEMS;
    const float* gt = tgt  + b * N_ITEMS;

    __builtin_prefetch(gp, 0, 1);                 // global_prefetch_b8
    __builtin_prefetch(gt, 0, 1);

#ifdef HAVE_TDM
    // --- real data path: Tensor Data Mover stages both rows into LDS -------
    if (tid < 32) {                               // wave 0 issues the DMA
        tdm_load_row_f32(gp, (unsigned)(uintptr_t)(&sm.p[0]), N_ITEMS);
        tdm_load_row_f32(gt, (unsigned)(uintptr_t)(&sm.t[0]), N_ITEMS);
#if __has_builtin(__builtin_amdgcn_s_wait_tensorcnt)
        __builtin_amdgcn_s_wait_tensorcnt(0);     // complete before barrier
#endif
    }
    __syncthreads();                              // TDM data visible to block
    for (int i = tid; i < N_ITEMS; i += NT_MAIN) {
        sm.pt[i] = make_float2(sm.p[i], sm.t[i]); // interleave in LDS
    }
#else
    for (int i = tid; i < N_ITEMS; i += NT_MAIN) {
        sm.pt[i] = make_float2(gp[i], gt[i]);
    }
#endif
    __syncthreads();

    // ---------------- ListNet KL + n_valid (slice 0 only) ------------------
    float kl = 0.0f, nv = 0.0f;
    if (s == 0) {
        float nvp = 0.0f, mpp = -3.0e38f, mtp = -3.0e38f;
        for (int i = tid; i < N_ITEMS; i += NT_MAIN) {
            float2 e = sm.pt[i];
            bool  v  = (e.y == e.y);
            float xp = v ? e.x : -1.0e30f;
            float xt = v ? e.y : -1.0e30f;
            nvp += v ? 1.0f : 0.0f;
            mpp = fmaxf(mpp, xp);
            mtp = fmaxf(mtp, xt);
        }
        nv        = block_sum(nvp, sm.sc0, tid);
        float mp  = block_max(mpp, sm.sc0, tid);
        float mt  = block_max(mtp, sm.sc0, tid);
        float spp = 0.0f, stp = 0.0f;
        for (int i = tid; i < N_ITEMS; i += NT_MAIN) {
            float2 e = sm.pt[i];
            bool  v  = (e.y == e.y);
            float xp = v ? e.x : -1.0e30f;
            float xt = v ? e.y : -1.0e30f;
            spp += __expf(xp - mp);               // invalid -> exp(-huge) == 0
            stp += __expf(xt - mt);
        }
        float sp  = block_sum(spp, sm.sc0, tid);
        float st  = block_sum(stp, sm.sc0, tid);
        float lsp = __logf(sp), lst = __logf(st);
        float klp = 0.0f;
        for (int i = tid; i < N_ITEMS; i += NT_MAIN) {
            float2 e = sm.pt[i];
            if (e.y == e.y) {
                float lt = (e.y - mt) - lst;      // log_softmax(target)
                float lp = (e.x - mp) - lsp;      // log_softmax(pred)
                klp += __expf(lt) * (lt - lp);    // tprob * (logt - logp)
            }
        }
        kl = block_sum(klp, sm.sc0, tid);
    }

    // ---------------- pair loop: Kendall + pairwise hinge ------------------
    // Branchless: m = (tj valid) ? 1 : 0; t_safe = valid ? tj : 0 so no NaN
    // reaches the arithmetic.  hinge(i,j)+hinge(j,i):
    //   td>0 -> 2*relu(1-pd); td<0 -> 2*relu(1+pd); td==0 -> relu(1-pd)+relu(1+pd)
    float kacc = 0.0f, pacc = 0.0f;
#pragma unroll
    for (int half = 0; half < 2; ++half) {
        int m = half ? (M_ROWS - 1 - tid) : tid;      // tid, 127-tid
        int i = s + N_SLICES * m;
        float2 ei = sm.pt[i];
        float ti = ei.y;
        if (ti == ti) {                               // row valid
            float pi = ei.x;
#pragma unroll 4
            for (int j = i + 1; j < N_ITEMS; ++j) {
                float2 e = sm.pt[j];                  // one ds_load_b64
                bool  vj = (e.y == e.y);
                float mk = vj ? 1.0f : 0.0f;
                float tj = vj ? e.y : 0.0f;
                float pd = pi - e.x;
                float td = ti - tj;
                kacc += mk * (fast_tanh(pd * 10.0f) * fast_tanh(td * 10.0f));
                float rm = fmaxf(1.0f - pd, 0.0f);
                float rp = fmaxf(1.0f + pd, 0.0f);
                float h  = (td > 0.0f) ? (rm + rm)
                         : ((td < 0.0f) ? (rp + rp) : (rm + rp));
                pacc += mk * h;
            }
        }
    }
    // Wave32 reduction on the matrix engine, then cross-wave via LDS.
    float kw = wave_sum_wmma(kacc);
    float pw = wave_sum_wmma(pacc);
    __syncthreads();
    if ((tid & 31) == 0) { sm.sc0[tid >> 5] = kw; sm.sc1[tid >> 5] = pw; }
    __syncthreads();
    if (tid == 0) {
        float knum = 0.0f, pnum = 0.0f;
#pragma unroll
        for (int k = 0; k < NWAVES; ++k) { knum += sm.sc0[k]; pnum += sm.sc1[k]; }
        float* o = ws + (s * N_BATCH + b) * 4;
        o[0] = knum; o[1] = pnum; o[2] = kl; o[3] = nv;
    }
}

// ----------------------------- finalize ------------------------------------
__global__ __launch_bounds__(32)
void combined_loss_finalize(const float* __restrict__ ws,
                            float* __restrict__ out) {
    int b = threadIdx.x;                 // one lane per batch, single wave32
    float knum = 0.0f, pnum = 0.0f;
#pragma unroll
    for (int s = 0; s < N_SLICES; ++s) {
        const float* o = ws + (s * N_BATCH + b) * 4;
        knum += o[0];
        pnum += o[1];
    }
    const float* o0 = ws + b * 4;        // slice 0 carries kl and n_valid
    float kl = o0[2], nv = o0[3];
    float contrib = 0.0f, ok = 0.0f;
    if (nv > 1.5f) {                     // sample_ok: n_valid > 1
        float d1 = 0.5f * nv * (nv - 1.0f);   // C(nv,2) concordance pairs
        float d2 = nv * (nv - 1.0f);          // ordered pairs
        contrib = -(knum / d1) + kl + (pnum / d2);
        ok = 1.0f;
    }
    float csum = wave_sum(contrib);
    float osum = wave_sum(ok);
    if (b == 0) out[0] = csum / fmaxf(osum, 1.0f);
}

// ----------------------------- launch --------------------------------------
extern "C" void kernel_launch(void* const* d_in, const int* in_sizes, int n_in,
                              void* d_out, int out_size, void* d_ws, size_t ws_size,
                              hipStream_t stream) {
    const float* pred = (const float*)d_in[0];
    const float* tgt  = (const float*)d_in[1];
    float* ws  = (float*)d_ws;
    float* out = (float*)d_out;
    combined_loss_main<<<dim3(N_BATCH, N_SLICES), dim3(NT_MAIN), 0, stream>>>(pred, tgt, ws);
    combined_loss_finalize<<<dim3(1), dim3(32), 0, stream>>>(ws, out);
}